// LCN_32134945308879
// MI455X (gfx1250) — compile-verified
//
#include <hip/hip_runtime.h>
#include <hip/hip_bf16.h>

// ---------------------------------------------------------------------------
// ST-GCN local graph convolution on MI455X (gfx1250), f32 via WMMA 16x16x4.
//
// out[n,o,v,t] = b[v,o] + sum_{k,c,kt} mask[v,k] *
//                W[v,o,c,k,kt] * x[n,c,idx[v,k], t+kt-1]   (zero-padded in t)
//
// v3: each wave owns a 32(row) x 64(col) output tile (8 f32 accumulators,
//     64 VGPRs) so every B fragment feeds 2 WMMAs and every A load feeds 4.
//     The (kt, chunk) loops are fully unrolled so all global/LDS loads use
//     compile-time immediate offsets from one base pointer per k — removing
//     the per-chunk address VALU and s_wait_xcnt serialization seen in v2.
// ---------------------------------------------------------------------------

typedef __attribute__((ext_vector_type(2))) float v2f;
typedef __attribute__((ext_vector_type(8))) float v8f;

#define V_NODES 25
#define KNB 3
#define CIN 64
#define COUT 64
#define NB 16
#define TLEN 512
#define TK 3
#define TT 128           // t-tile per block
#define XSTR 132         // t-extent in LDS: 130 valid + pad

__device__ __forceinline__ v8f wmma4(v2f a, v2f b, v8f c) {
    // D = A(16x4,f32) * B(4x16,f32) + C(16x16,f32)
    return __builtin_amdgcn_wmma_f32_16x16x4_f32(
        /*neg_a=*/false, a, /*neg_b=*/false, b,
        /*c_mod=*/(short)0, c, /*reuse_a=*/false, /*reuse_b=*/false);
}

// ---------------------------------------------------------------------------
// Pass 1: Wt[(((v*9 + k*3+kt)*32 + c/2)*64 + m)*2 + (c&1)]
//           = mask[v,k] ? W[v,m,c,k,kt] : 0
// Pair-interleaved over c so an A fragment is one 64-bit load.
// ---------------------------------------------------------------------------
__global__ void lcn_repack_w(const float* __restrict__ W,
                             const unsigned char* __restrict__ mask,
                             float* __restrict__ Wt) {
    int i = blockIdx.x * 256 + threadIdx.x;
    const int total = V_NODES * 9 * CIN * COUT;   // 921600
    if (i >= total) return;
    int e  = i & 1;
    int m  = (i >> 1) & 63;
    int p  = (i >> 7) & 31;
    int kk = (i >> 12) % 9;          // k*3 + kt
    int v  = i / (9 * CIN * COUT);
    int k  = kk / 3;
    int kt = kk % 3;
    int c  = p * 2 + e;
    float w = W[(((size_t)(v * COUT + m) * CIN + c) * KNB + k) * TK + kt];
    Wt[i] = mask[v * KNB + k] ? w : 0.0f;
}

// ---------------------------------------------------------------------------
// Pass 2: main WMMA kernel. grid = (T/TT, N, V), block = 128 (4 waves).
// Wave w: rows m0 = (w&1)*32 (2 tiles), cols cg = (w>>1)*64 (4 tiles).
// ---------------------------------------------------------------------------
__global__ void __launch_bounds__(128)
lcn_wmma_kernel(const float* __restrict__ x,
                const float* __restrict__ Wt,
                const float* __restrict__ bias,
                const int* __restrict__ gidx,
                float* __restrict__ out) {
    // xs[k][c/2][t][{c even, c odd}] : 3*32*132*2 floats = 101,376 B
    __shared__ float xs[KNB * 32 * XSTR * 2];

    const int t0 = blockIdx.x * TT;
    const int n  = blockIdx.y;
    const int v  = blockIdx.z;
    const int tid = threadIdx.x;

    // ---- Stage x[n, c, idx[v,k], t0-1 .. t0+TT] into LDS (zero halo) ------
    const float* xn = x + (size_t)n * CIN * V_NODES * TLEN;
    const int fill = KNB * CIN * (TT + 2);        // 24960 elements
    for (int i = tid; i < fill; i += 128) {
        int k = i / (CIN * (TT + 2));
        int r = i % (CIN * (TT + 2));
        int c = r / (TT + 2);
        int j = r % (TT + 2);                     // j=0 -> t0-1
        int t = t0 - 1 + j;
        int u = gidx[v * KNB + k];                // masked entries are 0 (safe)
        float val = 0.0f;
        if (t >= 0 && t < TLEN)
            val = xn[((size_t)c * V_NODES + u) * TLEN + t];
        xs[((k * 32 + (c >> 1)) * XSTR + j) * 2 + (c & 1)] = val;
    }
    __syncthreads();

    // ---- Per-wave WMMA reduction ------------------------------------------
    const int lane = tid & 31;
    const int wave = tid >> 5;            // 0..3
    const int m0   = (wave & 1) * 32;     // row tiles at m0, m0+16
    const int cg   = (wave >> 1) * 64;    // column tiles at cg+{0,16,32,48}
    const int half = lane >> 4;           // 0: K{0,1}, 1: K{2,3} of each chunk
    const int l    = lane & 15;           // row (A) / column (B) index

    v8f a00 = {}, a01 = {}, a02 = {}, a03 = {};   // rows m0..m0+15
    v8f a10 = {}, a11 = {}, a12 = {}, a13 = {};   // rows m0+16..m0+31

    const v2f* xsp = (const v2f*)xs;      // [(k*32+p)*XSTR + t] -> {c2p, c2p+1}

    for (int k = 0; k < KNB; ++k) {
        // Loop-invariant base pointers for this k; all inner offsets are
        // compile-time immediates after full unroll.
        const v2f* wb = (const v2f*)Wt
                        + ((size_t)v * 9 + k * TK) * (32 * COUT)
                        + half * COUT + m0 + l;
        const v2f* xb = xsp + (k * 32 + half) * XSTR + cg + l;
        __builtin_prefetch((const float*)(wb + 3 * 32 * COUT), 0, 1);
        #pragma unroll
        for (int kt = 0; kt < TK; ++kt) {
            const v2f* wp = wb + kt * (32 * COUT);
            const v2f* xr = xb + kt;
            #pragma unroll
            for (int i = 0; i < 16; ++i) {        // c4 = 4*i, p = 2*i + half
                v2f fa0 = wp[i * 2 * COUT];       // rows m0+l      (K pair)
                v2f fa1 = wp[i * 2 * COUT + 16];  // rows m0+16+l
                const v2f* xc = xr + i * 2 * XSTR;
                v2f b0 = xc[0];
                v2f b1 = xc[16];
                v2f b2 = xc[32];
                v2f b3 = xc[48];
                a00 = wmma4(fa0, b0, a00);
                a01 = wmma4(fa0, b1, a01);
                a02 = wmma4(fa0, b2, a02);
                a03 = wmma4(fa0, b3, a03);
                a10 = wmma4(fa1, b0, a10);
                a11 = wmma4(fa1, b1, a11);
                a12 = wmma4(fa1, b2, a12);
                a13 = wmma4(fa1, b3, a13);
            }
        }
    }

    // ---- Epilogue: bias + store -------------------------------------------
    // C/D layout: VGPR r, lanes 0-15 -> (M = m0+r,   N = l)
    //                     lanes 16-31 -> (M = m0+r+8, N = l)
    #pragma unroll
    for (int r = 0; r < 8; ++r) {
        const int mA = m0 + r + half * 8;
        const int mB = mA + 16;
        const float bvA = bias[v * COUT + mA];
        const float bvB = bias[v * COUT + mB];
        float* oA = out + ((size_t)(n * COUT + mA) * V_NODES + v) * TLEN + t0 + cg;
        float* oB = out + ((size_t)(n * COUT + mB) * V_NODES + v) * TLEN + t0 + cg;
        oA[l]      = a00[r] + bvA;
        oA[16 + l] = a01[r] + bvA;
        oA[32 + l] = a02[r] + bvA;
        oA[48 + l] = a03[r] + bvA;
        oB[l]      = a10[r] + bvB;
        oB[16 + l] = a11[r] + bvB;
        oB[32 + l] = a12[r] + bvB;
        oB[48 + l] = a13[r] + bvB;
    }
}

// ---------------------------------------------------------------------------
extern "C" void kernel_launch(void* const* d_in, const int* in_sizes, int n_in,
                              void* d_out, int out_size, void* d_ws, size_t ws_size,
                              hipStream_t stream) {
    const float*         x    = (const float*)d_in[0];
    const float*         W    = (const float*)d_in[1];
    const float*         bias = (const float*)d_in[2];
    const int*           gidx = (const int*)d_in[3];
    const unsigned char* mask = (const unsigned char*)d_in[4];
    float* out = (float*)d_out;
    float* Wt  = (float*)d_ws;   // needs 25*9*64*64*4 = 3,686,400 bytes

    const int total_w = V_NODES * 9 * CIN * COUT;
    lcn_repack_w<<<dim3((total_w + 255) / 256), dim3(256), 0, stream>>>(W, mask, Wt);

    lcn_wmma_kernel<<<dim3(TLEN / TT, NB, V_NODES), dim3(128), 0, stream>>>(
        x, Wt, bias, gidx, out);
}